// Points3DLoss_29523605193013
// MI455X (gfx1250) — compile-verified
//
#include <hip/hip_runtime.h>
#include <math.h>

typedef __attribute__((ext_vector_type(2))) float v2f;
typedef __attribute__((ext_vector_type(8))) float v8f;

#define N_OBS   4096
#define N_PRED  2048
#define BT      32
#define WAVES_PER_BLOCK 8
#define THREADS (WAVES_PER_BLOCK * 32)
#define BLOCKS_PER_SLICE (N_OBS / (16 * WAVES_PER_BLOCK))   // 32
#define N_PER_BATCH (16 * N_OBS)                            // 65536

// ---------------------------------------------------------------------------
// Phase 1: chamfer min-squared-distance, fully inside V_WMMA_F32_16X16X4_F32.
//
//   d(m,n) = |a_m|^2 + |b_n|^2 - 2 a_m.b_n
//
// Augmented K=4 product:  A row m = (-2x, -2y, -2z, 1)
//                         B col n = ( x',  y',  z', |b_n|^2)
//   => (A x B)[m][n] = |b_n|^2 - 2 a.b
// and |a_m|^2 rides in as the WMMA accumulator C (its VGPR layout —
// VGPR j: lanes 0-15 row j, lanes 16-31 row j+8 — matches naRow exactly).
// So D is the complete squared distance straight out of the matrix pipe.
//
// The clamp max(d,0) commutes with min, so it's applied once after the loop.
// B operand = one ds_load_b64 per lane per tile from pair-packed LDS.
// ---------------------------------------------------------------------------
__global__ __launch_bounds__(THREADS)
void chamfer_min_kernel(const float* __restrict__ obs,
                        const float* __restrict__ pred,
                        float* __restrict__ res) {
    // pk[n]        = (x, y)      -> B K=0,1 rows (consumed by lanes 0-15)
    // pk[NP + n]   = (z, |b|^2)  -> B K=2,3 rows (consumed by lanes 16-31)
    __shared__ v2f pk[2 * N_PRED];

    const int slice        = blockIdx.x / BLOCKS_PER_SLICE;   // 0..31 (= b*16+t)
    const int blockInSlice = blockIdx.x % BLOCKS_PER_SLICE;

    const float* predS = pred + (size_t)slice * N_PRED * 3;
    for (int i = threadIdx.x; i < N_PRED; i += THREADS) {
        float x = predS[i * 3 + 0];
        float y = predS[i * 3 + 1];
        float z = predS[i * 3 + 2];
        v2f lo, hi2;
        lo.x = x;  lo.y = y;
        hi2.x = z; hi2.y = x * x + y * y + z * z;
        pk[i] = lo;
        pk[N_PRED + i] = hi2;
    }
    __syncthreads();

    const int wave = threadIdx.x >> 5;
    const int lane = threadIdx.x & 31;
    const bool hi  = lane >= 16;                    // lanes 16..31 carry K=2,3
    const int obsTile = blockInSlice * WAVES_PER_BLOCK + wave;  // 0..255
    const int rowBase = obsTile * 16;
    const int m = rowBase + (lane & 15);

    // A operand (16x4 f32): lanes 0-15: {-2x, -2y}; lanes 16-31: {-2z, 1}.
    const float* op = obs + ((size_t)slice * N_OBS + m) * 3;
    __builtin_prefetch(op, 0, 0);                   // global_prefetch_b8
    float ox = op[0], oy = op[1], oz = op[2];
    float na = ox * ox + oy * oy + oz * oz;         // |a_m|^2, held per lane&15
    v2f A;
    A.x = -2.0f * (hi ? oz : ox);
    A.y = hi ? 1.0f : (-2.0f * oy);

    // C accumulator: C[j] = |a|^2 of row (j + 8*hi) — exactly the C/D layout.
    v8f Cacc;
#pragma unroll
    for (int j = 0; j < 8; ++j)
        Cacc[j] = __shfl(na, hi ? (j + 8) : j, 32);

    float mind[8];
#pragma unroll
    for (int j = 0; j < 8; ++j) mind[j] = 3.402823e+38f;

    const int nBase = (hi ? N_PRED : 0) + (lane & 15);

#pragma unroll 4
    for (int t = 0; t < N_PRED / 16; ++t) {
        v2f B = pk[nBase + t * 16];                 // one ds_load_b64

        v8f D = __builtin_amdgcn_wmma_f32_16x16x4_f32(
            /*neg_a=*/false, A, /*neg_b=*/false, B,
            /*c_mod=*/(short)0, Cacc, /*reuse_a=*/false, /*reuse_b=*/false);

#pragma unroll
        for (int j = 0; j < 8; ++j)
            mind[j] = fminf(mind[j], D[j]);         // full sq-dist already in D
    }

    // Min across the 16 lanes of each half-wave (columns of the tile).
#pragma unroll
    for (int j = 0; j < 8; ++j) {
        float v = mind[j];
        v = fminf(v, __shfl_xor(v, 1, 32));
        v = fminf(v, __shfl_xor(v, 2, 32));
        v = fminf(v, __shfl_xor(v, 4, 32));
        v = fminf(v, __shfl_xor(v, 8, 32));
        mind[j] = fmaxf(v, 0.0f);                   // deferred clamp
    }

    if ((lane & 15) == 0) {
        const size_t outBase = (size_t)slice * N_OBS + rowBase + (hi ? 8 : 0);
#pragma unroll
        for (int j = 0; j < 8; ++j)
            res[outBase + j] = sqrtf(mind[j]);
    }
}

// ---------------------------------------------------------------------------
// Phase 2: exact lower-median via MSB-first radix select on the (monotone)
// uint bit pattern of non-negative floats. One block per batch.
// mode 0: key = res[i]; mode 1: key = |res[i] - med[b]|.
// ---------------------------------------------------------------------------
__global__ __launch_bounds__(1024)
void radix_select_kernel(const float* __restrict__ res,
                         const float* __restrict__ med,
                         float* __restrict__ out,
                         int n, int k, int mode) {
    const int b = blockIdx.x;
    const float* data = res + (size_t)b * n;
    const float medv = mode ? med[b] : 0.0f;

    __shared__ unsigned int cnt;
    unsigned int prefix = 0;   // decided high bits (low bits zero)
    int kk = k;

    for (int bit = 31; bit >= 0; --bit) {
        if (threadIdx.x == 0) cnt = 0;
        __syncthreads();

        const unsigned int mask_hi =
            (bit == 31) ? 0u : (0xFFFFFFFFu << (bit + 1));
        unsigned int local = 0;
        for (int i = threadIdx.x; i < n; i += 1024) {
            float v = data[i];
            if (mode) v = fabsf(v - medv);
            unsigned int key = __float_as_uint(v);
            if ((key & mask_hi) == prefix && ((key >> bit) & 1u) == 0u)
                ++local;
        }
        atomicAdd(&cnt, local);
        __syncthreads();

        const unsigned int c0 = cnt;
        if ((unsigned int)kk >= c0) {
            kk -= (int)c0;
            prefix |= (1u << bit);
        }
        __syncthreads();
    }

    if (threadIdx.x == 0) out[b] = __uint_as_float(prefix);
}

// ---------------------------------------------------------------------------
// Phase 3: bisquare-weighted sum.
// ---------------------------------------------------------------------------
__global__ void init_out_kernel(float* out) { out[0] = 0.0f; }

__global__ __launch_bounds__(256)
void loss_kernel(const float* __restrict__ res,
                 const float* __restrict__ med,
                 const float* __restrict__ mad,
                 float* __restrict__ out) {
    const float TUNE = 4.6851f;
    const float MADC = 0.67449f;

    const int idx = blockIdx.x * blockDim.x + threadIdx.x;   // 0..131071
    const int b = idx / N_PER_BATCH;

    float r = res[idx];
    float stdv = mad[b] / MADC;
    float nr = r / (stdv * TUNE);
    float omn = 1.0f - nr * nr;
    float w = (nr >= 1.0f) ? 0.0f : (omn * omn);
    float contrib = 0.5f * w * r * r;

    // wave32 reduction, then one atomic per wave
    contrib += __shfl_xor(contrib, 1, 32);
    contrib += __shfl_xor(contrib, 2, 32);
    contrib += __shfl_xor(contrib, 4, 32);
    contrib += __shfl_xor(contrib, 8, 32);
    contrib += __shfl_xor(contrib, 16, 32);
    if ((threadIdx.x & 31) == 0)
        atomicAdd(out, contrib);
}

// ---------------------------------------------------------------------------
extern "C" void kernel_launch(void* const* d_in, const int* in_sizes, int n_in,
                              void* d_out, int out_size, void* d_ws, size_t ws_size,
                              hipStream_t stream) {
    const float* obs  = (const float*)d_in[0];   // (2,16,4096,3) f32
    const float* pred = (const float*)d_in[1];   // (2,16,2048,3) f32
    float* out = (float*)d_out;                  // scalar loss

    float* res = (float*)d_ws;                   // 131072 floats
    float* med = res + BT * N_OBS;               // 2 floats
    float* mad = med + 2;                        // 2 floats

    // 32 slices x 32 blocks, 8 waves/block -> one 16-row obs tile per wave
    chamfer_min_kernel<<<BT * BLOCKS_PER_SLICE, THREADS, 0, stream>>>(obs, pred, res);

    // lower median: rank (n-1)/2 = 32767 (0-indexed) of 65536
    radix_select_kernel<<<2, 1024, 0, stream>>>(res, nullptr, med,
                                                N_PER_BATCH, (N_PER_BATCH - 1) / 2, 0);
    radix_select_kernel<<<2, 1024, 0, stream>>>(res, med, mad,
                                                N_PER_BATCH, (N_PER_BATCH - 1) / 2, 1);

    init_out_kernel<<<1, 1, 0, stream>>>(out);
    loss_kernel<<<(2 * N_PER_BATCH) / 256, 256, 0, stream>>>(res, med, mad, out);
}